// ScaledRBFKernel_55052890800187
// MI455X (gfx1250) — compile-verified
//
#include <hip/hip_runtime.h>

typedef __attribute__((ext_vector_type(2))) float v2f;
typedef __attribute__((ext_vector_type(8))) float v8f;

// out[i,j] = outputscale * exp(-max(||x1_i - x2_j||^2, 0) / (2*l^2)),  l = 0.5
// Folded into a K=4 dot product computed by V_WMMA_F32_16X16X4_F32:
//   A_row(m) = [x1x, x1y, sq1, 1]
//   B_col(n) = [2c*x2x, 2c*x2y, -c, -c*sq2]       c = 2*log2(e)
//   A.B = c*(2*x1.x2 - sq1 - sq2) = -c*d2  ==>  out = scale * exp2(min(A.B, 0))
__global__ __launch_bounds__(256) void ScaledRBFKernel_wmma(
    const float* __restrict__ x1, const float* __restrict__ x2,
    const float* __restrict__ oscale, float* __restrict__ out,
    int tilesRow, int tilesCol, int M /* output row stride (cols) */) {
  const int lane = threadIdx.x & 31;
  const int warp = threadIdx.x >> 5;
  const int tile = blockIdx.x * 8 + warp;
  if (tile >= tilesRow * tilesCol) return;   // wave-uniform: EXEC stays all-1s

  const int tcol = tile % tilesCol;
  const int trow = tile / tilesCol;
  const int r = lane & 15;   // row of A / col of B within the tile
  const int h = lane >> 4;   // half: selects K-pair {0,1} vs {2,3}

  // Per-lane loads: row (trow*16+r) of x1, col (tcol*16+r) of x2
  const float2 a = ((const float2*)x1)[trow * 16 + r];
  const float2 b = ((const float2*)x2)[tcol * 16 + r];
  const float sq1 = a.x * a.x + a.y * a.y;
  const float sq2 = b.x * b.x + b.y * b.y;

  const float c    = 2.8853900817779268f;  // (1/(2*0.5^2)) * log2(e) = 2*log2e
  const float twoc = 2.0f * c;

  // A f32 16x4 layout: lane L holds (M=L%16, K = vgpr + 2*(L/16))
  // B f32 4x16 layout: lane L holds (K = vgpr + 2*(L/16), N=L%16)
  v2f A, B;
  A.x = h ? sq1       : a.x;
  A.y = h ? 1.0f      : a.y;
  B.x = h ? -c        : twoc * b.x;
  B.y = h ? -c * sq2  : twoc * b.y;

  v8f acc = {};
  // 8 args: (neg_a, A, neg_b, B, c_mod, C, reuse_a, reuse_b)
  v8f d = __builtin_amdgcn_wmma_f32_16x16x4_f32(
      false, A, false, B, (short)0, acc, false, false);

  const float s = *oscale;

  // D layout: lane L, VGPR v -> element (M = v + 8*(L/16), N = L%16)
  float* base = out + (size_t)(trow * 16 + 8 * h) * (size_t)M + (size_t)(tcol * 16 + r);
#pragma unroll
  for (int v = 0; v < 8; ++v) {
    float t = fminf(d[v], 0.0f);                 // clamp d2 >= 0
    float val = s * __builtin_amdgcn_exp2f(t);   // v_exp_f32
    __builtin_nontemporal_store(val, base + (size_t)v * (size_t)M);
  }
}

extern "C" void kernel_launch(void* const* d_in, const int* in_sizes, int n_in,
                              void* d_out, int out_size, void* d_ws, size_t ws_size,
                              hipStream_t stream) {
  const float* x1 = (const float*)d_in[0];   // [N,2] f32
  const float* x2 = (const float*)d_in[1];   // [M,2] f32
  const float* os = (const float*)d_in[2];   // scalar f32
  float* out = (float*)d_out;                // [N,M] f32

  const int n1 = in_sizes[0] / 2;            // 8192
  const int n2 = in_sizes[1] / 2;            // 8192
  const int tilesRow = n1 / 16;              // 512
  const int tilesCol = n2 / 16;              // 512
  const int totalTiles = tilesRow * tilesCol;
  const int blocks = (totalTiles + 7) / 8;   // 8 waves (tiles) per 256-thread block

  ScaledRBFKernel_wmma<<<blocks, 256, 0, stream>>>(x1, x2, os, out,
                                                   tilesRow, tilesCol, n2);
}